// TemporalNMFWithTime_18511309046027
// MI455X (gfx1250) — compile-verified
//
#include <hip/hip_runtime.h>
#include <hip/hip_bf16.h>
#include <math.h>

typedef __attribute__((ext_vector_type(2))) float v2f;
typedef __attribute__((ext_vector_type(8))) float v8f;

#define D_    32
#define T_    16
#define B_    128
#define F_    32
#define M_    4
#define E_    32
#define H_    128
#define NENT_ 4096
#define SLOPE_ 0.3f
#define TWO_PI_ 6.283185307179586f

#define XSTRIDE_ (H_ + 4)   // padded LDS row stride (floats): conflict-free, 8B-aligned

// ---------------------------------------------------------------- std(ages, ddof=1)
__global__ __launch_bounds__(256) void k_std(const float* __restrict__ ages,
                                             float* __restrict__ out) {
  __shared__ float ss[256], sq[256];
  float s = 0.f, q = 0.f;
  for (int i = threadIdx.x; i < D_ * NENT_; i += 256) {
    float v = ages[i];
    s += v; q += v * v;
  }
  ss[threadIdx.x] = s; sq[threadIdx.x] = q;
  __syncthreads();
  for (int o = 128; o > 0; o >>= 1) {
    if ((int)threadIdx.x < o) {
      ss[threadIdx.x] += ss[threadIdx.x + o];
      sq[threadIdx.x] += sq[threadIdx.x + o];
    }
    __syncthreads();
  }
  if (threadIdx.x == 0) {
    const float N = (float)(D_ * NENT_);
    float mean = ss[0] / N;
    float var  = (sq[0] - N * mean * mean) / (N - 1.0f);
    out[0] = sqrtf(var);
  }
}

// ---------------------------------------------------------------- a_clamped[d*B+b]
__global__ __launch_bounds__(256) void k_prep_a(const float* __restrict__ ages,
                                                const int* __restrict__ idxs,
                                                const float* __restrict__ stdp,
                                                float* __restrict__ a_out) {
  int i = blockIdx.x * 256 + threadIdx.x;          // i = d*B + b
  if (i >= D_ * B_) return;
  int d = i / B_, b = i % B_;
  float a = ages[d * NENT_ + idxs[b]] / stdp[0];
  a_out[i] = fmaxf(a, 0.0f);
}

// ---------------------------------------------------------------- norm-clipped embeddings
__global__ __launch_bounds__(128) void k_embs(const float* __restrict__ emb_table,
                                              const int* __restrict__ idxs,
                                              float* __restrict__ embs) {
  int b = threadIdx.x;
  if (b >= B_) return;
  const float* row = emb_table + (size_t)idxs[b] * E_;
  float sum = 0.f;
  for (int e = 0; e < E_; ++e) { float v = row[e]; sum += v * v; }
  float norm = sqrtf(sum);
  float scale = fminf(1.0f, 1.0f / (norm + 1e-7f));
  for (int e = 0; e < E_; ++e) embs[b * E_ + e] = row[e] * scale;
}

// ---------------------------------------------------------------- omr0[f]
__global__ __launch_bounds__(32) void k_omr0(const float* __restrict__ om,
                                             float* __restrict__ omr0) {
  int f = threadIdx.x;
  if (f >= F_) return;
  float sum = 0.f, r0 = 0.f;
  for (int m = 0; m < M_; ++m) {
    float sp = log1pf(expf(om[m * F_ + f]));
    if (m == 0) r0 = sp;
    sum += sp;
  }
  omr0[f] = r0 / sum * (float)F_;
}

// ---------------------------------------------------------------- off_mlp input rows: [embs(b), a_clamped(d,b)]
__global__ __launch_bounds__(256) void k_prep_offin(const float* __restrict__ embs,
                                                    const float* __restrict__ a_cl,
                                                    float* __restrict__ X) {
  int i = blockIdx.x * 256 + threadIdx.x;
  const int total = D_ * B_ * (E_ + 1);
  if (i >= total) return;
  int row = i / (E_ + 1), c = i % (E_ + 1);
  int b = row % B_;
  X[i] = (c < E_) ? embs[b * E_ + c] : a_cl[row];
}

// ---------------------------------------------------------------- time_mlp input rows: [f(d,b,:), sin t, cos t]
__global__ __launch_bounds__(256) void k_prep_time(const float* __restrict__ src,
                                                   float* __restrict__ X) {
  int i = blockIdx.x * 256 + threadIdx.x;
  const int CIN = F_ + 2;
  const int total = D_ * T_ * B_ * CIN;
  if (i >= total) return;
  int row = i / CIN, c = i % CIN;
  int b = row % B_;
  int t = (row / B_) % T_;
  int d = row / (B_ * T_);
  float v;
  if (c < F_) {
    v = src[(d * B_ + b) * F_ + c];
  } else {
    float tv = (float)t / (float)(T_ - 1) * TWO_PI_;
    v = (c == F_) ? sinf(tv) : cosf(tv);
  }
  X[i] = v;
}

// ---------------------------------------------------------------- generic WMMA fp32 MLP layer: Y = act(X @ W^T + b)
// 32 rows per block (two 16-row tiles, dual accumulators). 8 waves, wave w owns
// output columns [w*16, w*16+16). X tile staged in LDS, zero-padded to K%4,
// padded row stride to kill bank conflicts. Unguarded main K-loop + guarded tail.
__global__ __launch_bounds__(256) void k_mlp(const float* __restrict__ X,
                                             const float* __restrict__ W,
                                             const float* __restrict__ bias,
                                             float* __restrict__ Y,
                                             int Cin, int Cout, int leaky) {
  __shared__ float xs[32 * XSTRIDE_];                // 32 x 132 fp32 = 16.5 KB
  const int Cpad = (Cin + 3) & ~3;
  const int rowbase = blockIdx.x * 32;

  const int wave = threadIdx.x >> 5;
  const int lane = threadIdx.x & 31;

  for (int r = wave; r < 32; r += 8)
    for (int c = lane; c < Cpad; c += 32)
      xs[r * XSTRIDE_ + c] = (c < Cin) ? X[(size_t)(rowbase + r) * Cin + c] : 0.0f;
  __syncthreads();

  const int l15   = lane & 15;
  const int khalf = (lane >> 4) * 2;                  // lanes 16-31 hold K2/K3
  const int rhi   = (lane >> 4) * 8;                  // C frag: lanes 16-31 are M+8

  if (wave * 16 < Cout) {
    const int n = wave * 16 + l15;                    // output column
    const float* __restrict__ wrow = W + (size_t)n * Cin;
    const float* __restrict__ x0 = xs + l15 * XSTRIDE_;
    const float* __restrict__ x1 = xs + (16 + l15) * XSTRIDE_;

    v8f acc0 = {}, acc1 = {};
    int ks = 0;
    if ((Cin & 1) == 0) {
      // aligned fast path: one global_load_b64 per K-step for W
      for (; ks + 4 <= Cin; ks += 4) {
        int k = ks + khalf;
        v2f bw = *(const v2f*)(wrow + k);
        v2f a0, a1;
        a0.x = x0[k]; a0.y = x0[k + 1];
        a1.x = x1[k]; a1.y = x1[k + 1];
        acc0 = __builtin_amdgcn_wmma_f32_16x16x4_f32(false, a0, false, bw,
                                                     (short)0, acc0, false, false);
        acc1 = __builtin_amdgcn_wmma_f32_16x16x4_f32(false, a1, false, bw,
                                                     (short)0, acc1, false, false);
      }
    } else {
      for (; ks + 4 <= Cin; ks += 4) {
        int k = ks + khalf;
        v2f bw;
        bw.x = wrow[k];
        bw.y = wrow[k + 1];
        v2f a0, a1;
        a0.x = x0[k]; a0.y = x0[k + 1];
        a1.x = x1[k]; a1.y = x1[k + 1];
        acc0 = __builtin_amdgcn_wmma_f32_16x16x4_f32(false, a0, false, bw,
                                                     (short)0, acc0, false, false);
        acc1 = __builtin_amdgcn_wmma_f32_16x16x4_f32(false, a1, false, bw,
                                                     (short)0, acc1, false, false);
      }
    }
    if (ks < Cpad) {                                  // single guarded tail step
      int k = ks + khalf;
      v2f bw;
      bw.x = (k     < Cin) ? wrow[k]     : 0.0f;
      bw.y = (k + 1 < Cin) ? wrow[k + 1] : 0.0f;
      v2f a0, a1;
      a0.x = x0[k]; a0.y = x0[k + 1];
      a1.x = x1[k]; a1.y = x1[k + 1];
      acc0 = __builtin_amdgcn_wmma_f32_16x16x4_f32(false, a0, false, bw,
                                                   (short)0, acc0, false, false);
      acc1 = __builtin_amdgcn_wmma_f32_16x16x4_f32(false, a1, false, bw,
                                                   (short)0, acc1, false, false);
    }

    float bv = bias[n];
#pragma unroll
    for (int r = 0; r < 8; ++r) {
      float v0 = acc0[r] + bv;
      float v1 = acc1[r] + bv;
      if (leaky) {
        v0 = (v0 >= 0.f) ? v0 : SLOPE_ * v0;
        v1 = (v1 >= 0.f) ? v1 : SLOPE_ * v1;
      }
      Y[(size_t)(rowbase      + r + rhi) * Cout + n] = v0;
      Y[(size_t)(rowbase + 16 + r + rhi) * Cout + n] = v1;
    }
  }
}

// ---------------------------------------------------------------- femb = fage + factor_offsets
__global__ __launch_bounds__(256) void k_add(const float* __restrict__ a,
                                             const float* __restrict__ b,
                                             float* __restrict__ c, int n) {
  int i = blockIdx.x * 256 + threadIdx.x;
  if (i < n) c[i] = a[i] + b[i];
}

// ---------------------------------------------------------------- te = broadcast(f) + off
__global__ __launch_bounds__(256) void k_te(const float* __restrict__ src,
                                            const float* __restrict__ off,
                                            float* __restrict__ te) {
  int i = blockIdx.x * 256 + threadIdx.x;
  const int total = D_ * T_ * B_ * F_;
  if (i >= total) return;
  int f = i % F_;
  int row = i / F_;
  int b = row % B_;
  int d = row / (B_ * T_);
  te[i] = src[(d * B_ + b) * F_ + f] + off[i];
}

// ---------------------------------------------------------------- reconstruction: rec[d,t,i,j,m] = sum_k g_i g_j + db
// g[m][b][k] = relu(te[d,t,b,k*M+m]) * omr0[k*M+m];   G(128x8) @ G^T via WMMA f32 16x16x4.
#define GSTR_ 10   // padded k-stride (floats): conflict-free, 8B-aligned
__global__ __launch_bounds__(256) void k_recon(const float* __restrict__ te,
                                               const float* __restrict__ omr0,
                                               const float* __restrict__ daily_bias,
                                               float* __restrict__ rec) {
  __shared__ float g[M_ * B_ * GSTR_];               // 4*128*10 fp32 = 20 KB
  const int dt = blockIdx.x;                          // d*T + t
  const int d  = dt / T_;
  const float* tile = te + (size_t)dt * B_ * F_;

  for (int i = threadIdx.x; i < B_ * F_; i += 256) {
    int b = i / F_, f = i % F_;
    float v = fmaxf(tile[i], 0.0f) * omr0[f];
    g[((f % M_) * B_ + b) * GSTR_ + (f / M_)] = v;
  }
  __syncthreads();

  const int wave = threadIdx.x >> 5;
  const int lane = threadIdx.x & 31;
  const int l15  = lane & 15;
  const int khalf = (lane >> 4) * 2;
  const int rhi   = (lane >> 4) * 8;

  // 8 i-tiles x 8 j-tiles x 4 m = 256 jobs; two jobs in flight per wave
  for (int jb = wave; jb < 256; jb += 16) {
    int jobA = jb, jobB = jb + 8;
    int mA  = jobA & 3, jtA = (jobA >> 2) & 7, itA = jobA >> 5;
    int mB  = jobB & 3, jtB = (jobB >> 2) & 7, itB = jobB >> 5;
    const float* gA = g + mA * B_ * GSTR_;
    const float* gB = g + mB * B_ * GSTR_;
    v8f accA = {}, accB = {};
#pragma unroll
    for (int ks = 0; ks < F_ / M_; ks += 4) {
      int k = ks + khalf;
      v2f aA, bA, aB, bB;
      aA.x = gA[(itA * 16 + l15) * GSTR_ + k]; aA.y = gA[(itA * 16 + l15) * GSTR_ + k + 1];
      bA.x = gA[(jtA * 16 + l15) * GSTR_ + k]; bA.y = gA[(jtA * 16 + l15) * GSTR_ + k + 1];
      aB.x = gB[(itB * 16 + l15) * GSTR_ + k]; aB.y = gB[(itB * 16 + l15) * GSTR_ + k + 1];
      bB.x = gB[(jtB * 16 + l15) * GSTR_ + k]; bB.y = gB[(jtB * 16 + l15) * GSTR_ + k + 1];
      accA = __builtin_amdgcn_wmma_f32_16x16x4_f32(false, aA, false, bA,
                                                   (short)0, accA, false, false);
      accB = __builtin_amdgcn_wmma_f32_16x16x4_f32(false, aB, false, bB,
                                                   (short)0, accB, false, false);
    }
    float dbA = daily_bias[d * M_ + mA];
    float dbB = daily_bias[d * M_ + mB];
    int jA = jtA * 16 + l15;
    int jB = jtB * 16 + l15;
#pragma unroll
    for (int r = 0; r < 8; ++r) {
      int iA = itA * 16 + r + rhi;
      int iB = itB * 16 + r + rhi;
      rec[((((size_t)dt * B_ + iA) * B_ + jA) * M_) + mA] = accA[r] + dbA;
      rec[((((size_t)dt * B_ + iB) * B_ + jB) * M_) + mB] = accB[r] + dbB;
    }
  }
}

// =================================================================== launch
extern "C" void kernel_launch(void* const* d_in, const int* in_sizes, int n_in,
                              void* d_out, int out_size, void* d_ws, size_t ws_size,
                              hipStream_t stream) {
  const int*   idxs       = (const int*)  d_in[0];
  const float* ages       = (const float*)d_in[1];
  const float* emb_table  = (const float*)d_in[2];
  const float* daily_bias = (const float*)d_in[3];
  const float* output_map = (const float*)d_in[4];

  const float *ageW[5], *ageB[5], *offW[6], *offB[6], *timW[6], *timB[6];
  for (int i = 0; i < 5; ++i) { ageW[i] = (const float*)d_in[5  + 2*i]; ageB[i] = (const float*)d_in[6  + 2*i]; }
  for (int i = 0; i < 6; ++i) { offW[i] = (const float*)d_in[15 + 2*i]; offB[i] = (const float*)d_in[16 + 2*i]; }
  for (int i = 0; i < 6; ++i) { timW[i] = (const float*)d_in[27 + 2*i]; timB[i] = (const float*)d_in[28 + 2*i]; }

  // output layout (flattened return-tuple order)
  float* out = (float*)d_out;
  const size_t REC   = (size_t)D_ * T_ * B_ * B_ * M_;   // 33,554,432
  const size_t DBF   = (size_t)D_ * B_ * F_;             // 131,072
  const size_t DTBF  = (size_t)D_ * T_ * B_ * F_;        // 2,097,152
  float* rec_age  = out;
  float* rec_emb  = rec_age + REC;
  float* fage     = rec_emb + REC;
  float* femb     = fage + DBF;
  float* fage_te  = femb + DBF;
  float* femb_te  = fage_te + DTBF;
  float* fage_teo = femb_te + DTBF;
  float* femb_teo = fage_teo + DTBF;
  float* foff     = femb_teo + DTBF;
  float* embs     = foff + DBF;

  // workspace layout (floats)
  float* ws = (float*)d_ws;
  float* w_std   = ws;                                   // 1
  float* w_a     = ws + 64;                              // 4096
  float* w_omr   = w_a + D_ * B_;                        // 32
  float* w_xoff  = w_omr + 64;                           // 4096*33
  float* w_xtime = w_xoff + (size_t)D_ * B_ * (E_ + 1);  // 65536*34
  float* w_buf0  = w_xtime + (size_t)D_ * T_ * B_ * (F_ + 2);
  float* w_buf1  = w_buf0 + (size_t)D_ * T_ * B_ * H_;

  const int ROWS_S = D_ * B_;            // 4096
  const int ROWS_T = D_ * T_ * B_;       // 65536

  // ---- prep
  k_std   <<<1,   256, 0, stream>>>(ages, w_std);
  k_prep_a<<<(ROWS_S + 255) / 256, 256, 0, stream>>>(ages, idxs, w_std, w_a);
  k_embs  <<<1,   128, 0, stream>>>(emb_table, idxs, embs);
  k_omr0  <<<1,    32, 0, stream>>>(output_map, w_omr);
  {
    int total = ROWS_S * (E_ + 1);
    k_prep_offin<<<(total + 255) / 256, 256, 0, stream>>>(embs, w_a, w_xoff);
  }

  // ---- age_mlp: [1,H,H,H,H,F]  -> fage
  k_mlp<<<ROWS_S / 32, 256, 0, stream>>>(w_a,    ageW[0], ageB[0], w_buf0, 1,   H_, 1);
  k_mlp<<<ROWS_S / 32, 256, 0, stream>>>(w_buf0, ageW[1], ageB[1], w_buf1, H_,  H_, 1);
  k_mlp<<<ROWS_S / 32, 256, 0, stream>>>(w_buf1, ageW[2], ageB[2], w_buf0, H_,  H_, 1);
  k_mlp<<<ROWS_S / 32, 256, 0, stream>>>(w_buf0, ageW[3], ageB[3], w_buf1, H_,  H_, 1);
  k_mlp<<<ROWS_S / 32, 256, 0, stream>>>(w_buf1, ageW[4], ageB[4], fage,   H_,  F_, 0);

  // ---- off_mlp: [33,H,H,H,H,H,F] -> factor_offsets
  k_mlp<<<ROWS_S / 32, 256, 0, stream>>>(w_xoff, offW[0], offB[0], w_buf0, E_+1, H_, 1);
  k_mlp<<<ROWS_S / 32, 256, 0, stream>>>(w_buf0, offW[1], offB[1], w_buf1, H_,   H_, 1);
  k_mlp<<<ROWS_S / 32, 256, 0, stream>>>(w_buf1, offW[2], offB[2], w_buf0, H_,   H_, 1);
  k_mlp<<<ROWS_S / 32, 256, 0, stream>>>(w_buf0, offW[3], offB[3], w_buf1, H_,   H_, 1);
  k_mlp<<<ROWS_S / 32, 256, 0, stream>>>(w_buf1, offW[4], offB[4], w_buf0, H_,   H_, 1);
  k_mlp<<<ROWS_S / 32, 256, 0, stream>>>(w_buf0, offW[5], offB[5], foff,   H_,   F_, 0);

  // ---- femb = fage + factor_offsets
  k_add<<<((int)DBF + 255) / 256, 256, 0, stream>>>(fage, foff, femb, (int)DBF);

  // ---- timeembed + reconstruction, for fage then femb
  const float* srcs[2] = { fage,     femb     };
  float*       tes[2]  = { fage_te,  femb_te  };
  float*       teos[2] = { fage_teo, femb_teo };
  float*       recs[2] = { rec_age,  rec_emb  };
  for (int p = 0; p < 2; ++p) {
    int total = ROWS_T * (F_ + 2);
    k_prep_time<<<(total + 255) / 256, 256, 0, stream>>>(srcs[p], w_xtime);
    k_mlp<<<ROWS_T / 32, 256, 0, stream>>>(w_xtime, timW[0], timB[0], w_buf0, F_+2, H_, 1);
    k_mlp<<<ROWS_T / 32, 256, 0, stream>>>(w_buf0,  timW[1], timB[1], w_buf1, H_,   H_, 1);
    k_mlp<<<ROWS_T / 32, 256, 0, stream>>>(w_buf1,  timW[2], timB[2], w_buf0, H_,   H_, 1);
    k_mlp<<<ROWS_T / 32, 256, 0, stream>>>(w_buf0,  timW[3], timB[3], w_buf1, H_,   H_, 1);
    k_mlp<<<ROWS_T / 32, 256, 0, stream>>>(w_buf1,  timW[4], timB[4], w_buf0, H_,   H_, 1);
    k_mlp<<<ROWS_T / 32, 256, 0, stream>>>(w_buf0,  timW[5], timB[5], teos[p], H_,  F_, 0);
    k_te<<<((int)DTBF + 255) / 256, 256, 0, stream>>>(srcs[p], teos[p], tes[p]);
    k_recon<<<D_ * T_, 256, 0, stream>>>(tes[p], w_omr, daily_bias, recs[p]);
  }
}